// SEIModel_78099685310651
// MI455X (gfx1250) — compile-verified
//
#include <hip/hip_runtime.h>

typedef __attribute__((ext_vector_type(2))) float v2f;
typedef __attribute__((ext_vector_type(4))) float v4f;
typedef __attribute__((ext_vector_type(8))) float v8f;

// ---------------------------------------------------------------------------
// Kernel 1: one wave (32 lanes). Computes L@S0 and L@I0 with
// V_WMMA_F32_16X16X4_F32 (L padded 13->16, B = [S0 | I0 | 0...] as 4x16
// fragments, 4 K-chunks), then the per-row Euler step constants:
//   cS[r] = dt * ( -K*(L@S0)[r] - beta*E0[r]*S0[r] - gamma*I0[r]*S0[r] )
//   cI[r] = dt * ( -Ki*(L@I0)[r] + alpha*E0[r] - mu*I0[r] )
// written to ws[0..12] and ws[13..25].
// ---------------------------------------------------------------------------
__launch_bounds__(32)
__global__ void sei_prepare_kernel(const float* __restrict__ S0,
                                   const float* __restrict__ E0,
                                   const float* __restrict__ I0,
                                   const float* __restrict__ L,
                                   const float* __restrict__ beta_p,
                                   const float* __restrict__ gamma_p,
                                   const float* __restrict__ K_p,
                                   const float* __restrict__ Ki_p,
                                   const float* __restrict__ dt_p,
                                   float* __restrict__ cws) {
    const int lane  = threadIdx.x;   // 0..31, EXEC all-1s through the WMMAs
    const int m     = lane & 15;     // A: row index / B: column index (N)
    const int khalf = lane >> 4;     // 0 -> K pair {0,1}, 1 -> K pair {2,3}

    __shared__ float LS[16];   // (L@S0) padded
    __shared__ float LI[16];   // (L@I0) padded

    v8f acc = {};
    #pragma unroll
    for (int kb = 0; kb < 4; ++kb) {
        const int k0 = kb * 4 + khalf * 2;   // this lane's first K column
        // A fragment: 16x4 tile of padded L (row m, cols k0, k0+1)
        v2f a;
        a.x = (m < 13 && (k0 + 0) < 13) ? L[m * 13 + k0 + 0] : 0.0f;
        a.y = (m < 13 && (k0 + 1) < 13) ? L[m * 13 + k0 + 1] : 0.0f;
        // B fragment: 4x16, B[k][0]=S0[k], B[k][1]=I0[k], else 0
        const float sk0 = ((k0 + 0) < 13) ? S0[k0 + 0] : 0.0f;
        const float sk1 = ((k0 + 1) < 13) ? S0[k0 + 1] : 0.0f;
        const float ik0 = ((k0 + 0) < 13) ? I0[k0 + 0] : 0.0f;
        const float ik1 = ((k0 + 1) < 13) ? I0[k0 + 1] : 0.0f;
        v2f b;
        b.x = (m == 0) ? sk0 : ((m == 1) ? ik0 : 0.0f);
        b.y = (m == 0) ? sk1 : ((m == 1) ? ik1 : 0.0f);
        // 8 args: (neg_a, A, neg_b, B, c_mod, C, reuse_a, reuse_b)
        acc = __builtin_amdgcn_wmma_f32_16x16x4_f32(
            false, a, false, b, (short)0, acc, false, false);
    }

    // D layout (16x16 f32): VGPR j -> lanes 0-15: M=j, N=lane;
    //                                 lanes 16-31: M=8+j, N=lane-16.
    // Column 0 (L@S0): lanes 0 & 16. Column 1 (L@I0): lanes 1 & 17.
    if (lane == 0) {
        #pragma unroll
        for (int j = 0; j < 8; ++j) LS[j] = acc[j];
    }
    if (lane == 16) {
        #pragma unroll
        for (int j = 0; j < 8; ++j) LS[8 + j] = acc[j];
    }
    if (lane == 1) {
        #pragma unroll
        for (int j = 0; j < 8; ++j) LI[j] = acc[j];
    }
    if (lane == 17) {
        #pragma unroll
        for (int j = 0; j < 8; ++j) LI[8 + j] = acc[j];
    }
    __syncthreads();

    if (lane < 13) {
        const float beta  = *beta_p;
        const float gamma = *gamma_p;
        const float K     = *K_p;
        const float Ki    = *Ki_p;
        const float dt    = *dt_p;
        const float alpha = 0.2f;
        const float mu    = 0.01f;
        const float s0 = S0[lane], e0 = E0[lane], i0 = I0[lane];
        const float dS = -K  * LS[lane] - beta * e0 * s0 - gamma * i0 * s0;
        const float dI = -Ki * LI[lane] + alpha * e0 - mu * i0;
        cws[lane]      = dt * dS;
        cws[13 + lane] = dt * dI;
    }
}

// ---------------------------------------------------------------------------
// Kernel 2: bandwidth-bound fill. out[m][r][t] = fma(t, c[m*13+r], x0[r]).
// float4 stores (global_store_b128); nt here is divisible by 4 so each
// float4 stays inside one row.
// ---------------------------------------------------------------------------
__launch_bounds__(256)
__global__ void sei_fill_kernel(const float* __restrict__ S0,
                                const float* __restrict__ I0,
                                const float* __restrict__ cws,
                                float* __restrict__ out,
                                int nt, int total4) {
    const int i4 = blockIdx.x * blockDim.x + threadIdx.x;
    if (i4 >= total4) return;
    const int e      = i4 << 2;          // element index, < 26*nt (fits int)
    const int rowAll = e / nt;           // 0..25
    const int t      = e - rowAll * nt;
    const int r      = (rowAll < 13) ? rowAll : rowAll - 13;
    const float c    = cws[rowAll];
    const float x0   = (rowAll < 13) ? S0[r] : I0[r];
    const float tf   = (float)t;         // t < 2^24, exact in fp32
    v4f val;
    val.x = fmaf(tf,        c, x0);
    val.y = fmaf(tf + 1.0f, c, x0);
    val.z = fmaf(tf + 2.0f, c, x0);
    val.w = fmaf(tf + 3.0f, c, x0);
    *(v4f*)(out + e) = val;
}

// Scalar fallback if nt % 4 != 0 (not the case for the given harness).
__launch_bounds__(256)
__global__ void sei_fill_scalar(const float* __restrict__ S0,
                                const float* __restrict__ I0,
                                const float* __restrict__ cws,
                                float* __restrict__ out,
                                int nt, int total) {
    const int e = blockIdx.x * blockDim.x + threadIdx.x;
    if (e >= total) return;
    const int rowAll = e / nt;
    const int t      = e - rowAll * nt;
    const int r      = (rowAll < 13) ? rowAll : rowAll - 13;
    const float x0   = (rowAll < 13) ? S0[r] : I0[r];
    out[e] = fmaf((float)t, cws[rowAll], x0);
}

extern "C" void kernel_launch(void* const* d_in, const int* in_sizes, int n_in,
                              void* d_out, int out_size, void* d_ws, size_t ws_size,
                              hipStream_t stream) {
    const float* S0    = (const float*)d_in[0];
    const float* E0    = (const float*)d_in[1];
    const float* I0    = (const float*)d_in[2];
    const float* L     = (const float*)d_in[3];
    const float* beta  = (const float*)d_in[4];
    const float* gamma_= (const float*)d_in[5];
    const float* K     = (const float*)d_in[6];
    const float* Ki    = (const float*)d_in[7];
    const float* dt    = (const float*)d_in[8];
    // d_in[9] is nt on device; recover it from the host-visible out_size.
    const int nt = out_size / 26;                 // out is [2,13,nt]
    float* cws = (float*)d_ws;                    // 26 floats of scratch
    float* out = (float*)d_out;

    sei_prepare_kernel<<<1, 32, 0, stream>>>(S0, E0, I0, L, beta, gamma_,
                                             K, Ki, dt, cws);

    if ((nt & 3) == 0) {
        const int total4 = out_size >> 2;         // 26*nt/4 float4 stores
        const int blocks = (total4 + 255) / 256;
        sei_fill_kernel<<<blocks, 256, 0, stream>>>(S0, I0, cws, out, nt, total4);
    } else {
        const int blocks = (out_size + 255) / 256;
        sei_fill_scalar<<<blocks, 256, 0, stream>>>(S0, I0, cws, out, nt, out_size);
    }
}